// SimpleLoss_32238024523892
// MI455X (gfx1250) — compile-verified
//
#include <hip/hip_runtime.h>

typedef float v2f __attribute__((ext_vector_type(2)));
typedef float v8f __attribute__((ext_vector_type(8)));

#define B_ 16
#define T_ 20
#define H_ 512
#define W_ 512
#define N_ 1024
#define P_ 20

// ws layout: float[0..319]  = cv_last[b][p]   (value at trajectory n = N-1)
//            uint [320..335] = worst[b] (float bits, monotone for >=0 floats)

__global__ void sl_init_kernel(const float* __restrict__ cv,
                               const long long* __restrict__ traj,
                               float* __restrict__ ws) {
    int tid = threadIdx.x;
    if (tid < B_ * P_) {
        int b = tid / P_;
        int p = tid % P_;
        long long base = (((long long)b * N_ + (N_ - 1)) * P_ + p) * 3;
        long long t = traj[base + 0];
        long long h = traj[base + 1];
        long long w = traj[base + 2];
        long long idx = (((long long)b * T_ + t) * H_ + h) * W_ + w;
        ws[tid] = cv[idx];
    } else if (tid < B_ * P_ + B_) {
        ((unsigned*)ws)[tid] = 0u;   // worst[b] = 0.0f bits
    }
}

// Each wave (32 lanes) handles 16 consecutive trajectories of one batch.
// Margins are placed in V_WMMA_F32_16X16X4_F32 A-matrix layout:
//   lane L: row M = L & 15, K-half = (L>>4)*2; a.x -> K=kh, a.y -> K=kh+1
// B = all-ones (4x16), so D[i][j] = sum_k A[i][k] + C[i][j]: every column of D
// holds the per-trajectory margin sum. 5 chunks of K=4 cover P=20 exactly.
__global__ void sl_margin_kernel(const float* __restrict__ cv,
                                 const long long* __restrict__ traj,
                                 const float* __restrict__ dist_p,
                                 float* __restrict__ ws) {
    int lane = threadIdx.x & 31;
    int wave = (int)((blockIdx.x * blockDim.x + threadIdx.x) >> 5);  // 0..1023
    int b  = wave >> 6;          // 64 waves per batch
    int n0 = (wave & 63) << 4;   // 16 trajectories per wave
    int row = lane & 15;
    int kh  = (lane >> 4) << 1;  // 0 or 2
    int n = n0 + row;

    float dist = *dist_p;
    const float* cvlast = ws + b * P_;
    long long tb = ((long long)b * N_ + n) * P_ * 3;

    v2f ones;
    ones.x = 1.0f;
    ones.y = 1.0f;
    v8f acc = {};

#pragma unroll
    for (int c = 0; c < 5; ++c) {
        int p0 = 4 * c + kh;
        v2f a;
        {
            long long base = tb + (long long)p0 * 3;
            long long t = traj[base + 0];
            long long h = traj[base + 1];
            long long w = traj[base + 2];
            float v = cv[(((long long)b * T_ + t) * H_ + h) * W_ + w];
            float m = cvlast[p0] - v + dist;
            a.x = m > 0.0f ? m : 0.0f;
        }
        {
            long long base = tb + (long long)(p0 + 1) * 3;
            long long t = traj[base + 0];
            long long h = traj[base + 1];
            long long w = traj[base + 2];
            float v = cv[(((long long)b * T_ + t) * H_ + h) * W_ + w];
            float m = cvlast[p0 + 1] - v + dist;
            a.y = m > 0.0f ? m : 0.0f;
        }
        // D = A x ones + C : accumulates per-row sums across the K=4 chunk
        acc = __builtin_amdgcn_wmma_f32_16x16x4_f32(
            /*neg_a=*/false, a, /*neg_b=*/false, ones,
            /*c_mod=*/(short)0, acc, /*reuse_a=*/false, /*reuse_b=*/false);
    }

    // Lane L holds rows (0..7)+8*(L>=16) in acc[0..7], identical across columns.
    float m = acc[0];
#pragma unroll
    for (int r = 1; r < 8; ++r) m = fmaxf(m, acc[r]);

    if (lane == 0 || lane == 16) {
        unsigned* worst = (unsigned*)ws + B_ * P_;
        atomicMax(&worst[b], __float_as_uint(m));  // margins >= 0 -> uint order ok
    }
}

__global__ void sl_final_kernel(const float* __restrict__ ws, float* __restrict__ out) {
    if (threadIdx.x == 0) {
        const unsigned* worst = (const unsigned*)ws + B_ * P_;
        float s = 0.0f;
        for (int b = 0; b < B_; ++b) s += __uint_as_float(worst[b]);
        out[0] = s;
    }
}

extern "C" void kernel_launch(void* const* d_in, const int* in_sizes, int n_in,
                              void* d_out, int out_size, void* d_ws, size_t ws_size,
                              hipStream_t stream) {
    const float*     cv   = (const float*)d_in[0];
    const long long* traj = (const long long*)d_in[1];
    const float*     dist = (const float*)d_in[2];
    float* ws  = (float*)d_ws;
    float* out = (float*)d_out;

    (void)in_sizes; (void)n_in; (void)out_size; (void)ws_size;

    sl_init_kernel<<<1, 384, 0, stream>>>(cv, traj, ws);

    // 1024 waves total: 128 blocks x 256 threads (8 wave32 per block)
    sl_margin_kernel<<<128, 256, 0, stream>>>(cv, traj, dist, ws);

    sl_final_kernel<<<1, 32, 0, stream>>>(ws, out);
}